// YoloLoss_47227460387532
// MI455X (gfx1250) — compile-verified
//
#include <hip/hip_runtime.h>

// ---------------------------------------------------------------------------
// YOLOv1 loss on MI455X (gfx1250): memory-bound streaming reduction.
// 192.7 MB read / 23.3 TB/s -> ~8.3 us floor. Two cells per thread so all
// loads are 16B-aligned global_load_b128. Deterministic two-kernel sum.
// Wave reduction = one V_WMMA_F32_16X16X4_F32 (A = partials, B = ones):
// D[m][n] = p[m] + p[m+16], so per-lane sum of the 8 D-VGPRs yields the
// low-rows half-sum in lanes 0-15 and the high-rows half-sum in lanes 16-31.
// No cross-lane shuffles needed at all.
// ---------------------------------------------------------------------------

#define SGRID 14

typedef float v2f __attribute__((ext_vector_type(2)));
typedef float v8f __attribute__((ext_vector_type(8)));

// 256-thread block reduction. Result valid on thread 0. Deterministic order.
// Must be entered with EXEC == all ones (WMMA requirement).
__device__ __forceinline__ float blockReduceSum(float p, float* lds) {
#if __has_builtin(__builtin_amdgcn_wmma_f32_16x16x4_f32)
    v2f a; a[0] = p;    a[1] = 0.0f;   // A[m][0]=p[m], A[m][2]=p[m+16]
    v2f b; b[0] = 1.0f; b[1] = 1.0f;   // B = ones(4x16)
    v8f c = {};
    c = __builtin_amdgcn_wmma_f32_16x16x4_f32(
        /*neg_a=*/false, a, /*neg_b=*/false, b,
        /*c_mod=*/(short)0, c, /*reuse_a=*/false, /*reuse_b=*/false);
    // lane L<16: sum of rows 0..7 ; lane L>=16: sum of rows 8..15
    const float s = ((c[0] + c[1]) + (c[2] + c[3])) +
                    ((c[4] + c[5]) + (c[6] + c[7]));
    if ((threadIdx.x & 15) == 0) lds[threadIdx.x >> 4] = s;  // lanes 0 and 16
    __syncthreads();
    float tot = 0.0f;
    if (threadIdx.x == 0) {
        const int ne = blockDim.x >> 4;                       // 16 entries
        for (int i = 0; i < ne; ++i) tot += lds[i];           // fixed order
    }
    return tot;
#else
    for (int m = 16; m; m >>= 1) p += __shfl_xor(p, m, 32);
    if ((threadIdx.x & 31) == 0) lds[threadIdx.x >> 5] = p;
    __syncthreads();
    float tot = 0.0f;
    if (threadIdx.x == 0) {
        const int ne = blockDim.x >> 5;
        for (int i = 0; i < ne; ++i) tot += lds[i];
    }
    return tot;
#endif
}

// Per-cell YOLOv1 loss; pv/tv point at 30 register-resident floats.
__device__ __forceinline__ float cellLoss(const float* pv, const float* tv) {
    const float invS = 1.0f / (float)SGRID;

    const float obj   = (tv[4] == 1.0f) ? 1.0f : 0.0f;
    const float noobj = (tv[4] == 0.0f) ? 1.0f : 0.0f;

    // no-object confidence loss (channels 4 and 9)
    const float d4 = pv[4] - tv[4];
    const float d9 = pv[9] - tv[9];
    const float noobj_loss = noobj * (d4 * d4 + d9 * d9);

    // class loss over channels 10..29
    float cls = 0.0f;
#pragma unroll
    for (int i = 10; i < 30; ++i) {
        const float d = pv[i] - tv[i];
        cls += d * d;
    }
    cls *= obj;

    // IoU for both box pairs
    float iou[2];
#pragma unroll
    for (int bx = 0; bx < 2; ++bx) {
        const float* pb = pv + 5 * bx;
        const float* tb = tv + 5 * bx;
        const float pcx = pb[0] * invS, pcy = pb[1] * invS, pw = pb[2], ph = pb[3];
        const float tcx = tb[0] * invS, tcy = tb[1] * invS, tw = tb[2], th = tb[3];
        const float px0 = pcx - 0.5f * pw, py0 = pcy - 0.5f * ph;
        const float px1 = pcx + 0.5f * pw, py1 = pcy + 0.5f * ph;
        const float tx0 = tcx - 0.5f * tw, ty0 = tcy - 0.5f * th;
        const float tx1 = tcx + 0.5f * tw, ty1 = tcy + 0.5f * th;
        const float lx = fmaxf(px0, tx0), ly = fmaxf(py0, ty0);
        const float rx = fminf(px1, tx1), ry = fminf(py1, ty1);
        const float iw = fmaxf(rx - lx, 0.0f), ih = fmaxf(ry - ly, 0.0f);
        const float inter = iw * ih;
        const float ap = (px1 - px0) * (py1 - py0);
        const float at = (tx1 - tx0) * (ty1 - ty0);
        const float den = ap + at - inter;
        iou[bx] = (den > 0.0f) ? (inter / den) : 0.0f;
    }

    // responsible box: box 1 wins ties
    const bool  pick1 = iou[0] <= iou[1];
    const float ciou  = pick1 ? iou[1] : iou[0];

    const float bpx = pick1 ? pv[5] : pv[0];
    const float bpy = pick1 ? pv[6] : pv[1];
    const float bpw = pick1 ? pv[7] : pv[2];
    const float bph = pick1 ? pv[8] : pv[3];
    const float bpc = pick1 ? pv[9] : pv[4];
    const float btx = pick1 ? tv[5] : tv[0];
    const float bty = pick1 ? tv[6] : tv[1];
    const float btw = pick1 ? tv[7] : tv[2];
    const float bth = pick1 ? tv[8] : tv[3];

    const float dx = bpx - btx;
    const float dy = bpy - bty;
    const float dw = sqrtf(bpw) - sqrtf(btw);
    const float dh = sqrtf(bph) - sqrtf(bth);
    const float reg = obj * (dx * dx + dy * dy + dw * dw + dh * dh);

    const float dconf   = bpc - ciou;
    const float contain = obj * dconf * dconf;

    return 5.0f * reg + contain + 0.5f * noobj_loss + cls;
}

// One PAIR of cells (240 B, 16B-aligned) per thread: 15x global_load_b128
// per tensor, fully coalesced streaming.
__global__ void __launch_bounds__(256)
yolo_partial(const float* __restrict__ pred, const float* __restrict__ tgt,
             float* __restrict__ partials, int totalPairs) {
    __shared__ float lds[16];
    const int stride = gridDim.x * blockDim.x;
    float acc = 0.0f;

    for (int pc = blockIdx.x * blockDim.x + threadIdx.x; pc < totalPairs; pc += stride) {
        const float4* pp = reinterpret_cast<const float4*>(pred + (size_t)pc * 60);
        const float4* tp = reinterpret_cast<const float4*>(tgt  + (size_t)pc * 60);

        float pv[60], tv[60];
#pragma unroll
        for (int j = 0; j < 15; ++j) {            // b128 loads
            const float4 v = pp[j];
            const float4 u = tp[j];
            pv[4 * j + 0] = v.x; pv[4 * j + 1] = v.y;
            pv[4 * j + 2] = v.z; pv[4 * j + 3] = v.w;
            tv[4 * j + 0] = u.x; tv[4 * j + 1] = u.y;
            tv[4 * j + 2] = u.z; tv[4 * j + 3] = u.w;
        }
        acc += cellLoss(pv, tv) + cellLoss(pv + 30, tv + 30);
    }

    // all lanes reconverged -> EXEC all ones for WMMA
    const float tot = blockReduceSum(acc, lds);
    if (threadIdx.x == 0) partials[blockIdx.x] = tot;
}

__global__ void __launch_bounds__(256)
yolo_final(const float* __restrict__ partials, float* __restrict__ out,
           int n, float invN) {
    __shared__ float lds[16];
    float acc = 0.0f;
    for (int i = threadIdx.x; i < n; i += 256) acc += partials[i];  // fixed order
    const float tot = blockReduceSum(acc, lds);
    if (threadIdx.x == 0) out[0] = tot * invN;
}

extern "C" void kernel_launch(void* const* d_in, const int* in_sizes, int n_in,
                              void* d_out, int out_size, void* d_ws, size_t ws_size,
                              hipStream_t stream) {
    const float* pred = (const float*)d_in[0];
    const float* tgt  = (const float*)d_in[1];
    float* out        = (float*)d_out;
    float* partials   = (float*)d_ws;

    const int totalCells = in_sizes[0] / 30;                  // N*S*S = 802816
    const int N          = totalCells / (SGRID * SGRID);      // batch = 4096
    const int totalPairs = totalCells / 2;                    // 401408
    const int NT = 256;
    const int NB = (totalPairs + NT - 1) / NT;                // 1568: exact fit

    yolo_partial<<<NB, NT, 0, stream>>>(pred, tgt, partials, totalPairs);
    yolo_final<<<1, NT, 0, stream>>>(partials, out, NB, 1.0f / (float)N);
}